// GeneratePrediction_24137716204221
// MI455X (gfx1250) — compile-verified
//
#include <hip/hip_runtime.h>
#include <hip/hip_bf16.h>
#include <limits.h>

#define B_   512
#define K_   10
#define S_   40
#define VSC  7000
#define VCO  1000

// ---------- CDNA5 async global->LDS path (guarded; fallback = plain loads) ----
#if defined(__AMDGCN__) && __has_builtin(__builtin_amdgcn_global_load_async_to_lds_b128)
#define USE_ASYNC_LDS 1
#else
#define USE_ASYNC_LDS 0
#endif

typedef int v4i_ __attribute__((ext_vector_type(4)));
#define AS1_ __attribute__((address_space(1)))
#define AS3_ __attribute__((address_space(3)))

__device__ __forceinline__ void wait_async_zero() {
#if defined(__AMDGCN__)
#if __has_builtin(__builtin_amdgcn_s_wait_asynccnt)
    __builtin_amdgcn_s_wait_asynccnt(0);
#else
    asm volatile("s_wait_asynccnt 0" ::: "memory");
#endif
#endif
}

__device__ __forceinline__ void combine_max(float& bv, int& bi, float v, int i) {
    // larger value wins; on exact tie, lower index wins (matches jax.lax.top_k)
    if (v > bv || (v == bv && i < bi)) { bv = v; bi = i; }
}

// ---------------------------------------------------------------------------
// Kernel 1: top-10 over 7000 logits per row (B=512 rows).
// Row cached in LDS (async b128 copies), then 10 rounds of masked argmax.
// ---------------------------------------------------------------------------
__global__ __launch_bounds__(256) void topk_sc_kernel(const float* __restrict__ logits,
                                                      int* __restrict__ out_labels) {
    __shared__ float vals[VSC];
    __shared__ float rv[8];
    __shared__ int   ri[8];

    const int b    = blockIdx.x;
    const int tid  = threadIdx.x;
    const int lane = tid & 31;
    const int wid  = tid >> 5;
    const float* row = logits + (size_t)b * VSC;
    const float NEG  = -__builtin_inff();

#if USE_ASYNC_LDS
    // 7000 floats = 1750 x b128 chunks; each lane issues async copies.
    for (int c = tid; c < VSC / 4; c += 256) {
        AS1_ v4i_* g = (AS1_ v4i_*)(row + c * 4);
        AS3_ v4i_* l = (AS3_ v4i_*)(&vals[c * 4]);
        __builtin_amdgcn_global_load_async_to_lds_b128(g, l, 0, 0);
    }
    wait_async_zero();
#else
    for (int i = tid; i < VSC; i += 256) vals[i] = row[i];
#endif
    __syncthreads();

    for (int r = 0; r < K_; ++r) {
        float bv = NEG;
        int   bi = 0x7FFFFFFF;
        for (int i = tid; i < VSC; i += 256) combine_max(bv, bi, vals[i], i);

#pragma unroll
        for (int m = 16; m; m >>= 1) {
            float ov = __shfl_xor(bv, m, 32);
            int   oi = __shfl_xor(bi, m, 32);
            combine_max(bv, bi, ov, oi);
        }
        if (lane == 0) { rv[wid] = bv; ri[wid] = bi; }
        __syncthreads();

        if (wid == 0) {
            float v2 = (lane < 8) ? rv[lane] : NEG;
            int   i2 = (lane < 8) ? ri[lane] : 0x7FFFFFFF;
#pragma unroll
            for (int m = 4; m; m >>= 1) {
                float ov = __shfl_xor(v2, m, 32);
                int   oi = __shfl_xor(i2, m, 32);
                combine_max(v2, i2, ov, oi);
            }
            if (lane == 0) {
                out_labels[b * K_ + r] = i2;
                vals[i2] = NEG;          // mask out for next round
            }
        }
        __syncthreads();
    }
}

// ---------------------------------------------------------------------------
// Kernel 2: beam search degenerates to greedy argmax over V=1000 per (b,s).
// One wave per row; rows 0..20479 = lr, 20480..40959 = ul.
// Lanes 0..9 replicate the token into the 10 beam slots.
// ---------------------------------------------------------------------------
__global__ __launch_bounds__(256) void argmax_seq_kernel(const float* __restrict__ lr,
                                                         const float* __restrict__ ul,
                                                         int* __restrict__ out_lr,
                                                         int* __restrict__ out_ul) {
    const int lane = threadIdx.x & 31;
    const int wid  = threadIdx.x >> 5;
    const int rowg = blockIdx.x * 8 + wid;        // 0 .. 2*B*S-1
    const int half = rowg / (B_ * S_);            // 0 = lr, 1 = ul
    const int r    = rowg - half * (B_ * S_);     // 0 .. B*S-1
    const int b    = r / S_;
    const int s    = r - b * S_;

    const float4* src = (const float4*)((half ? ul : lr) + (size_t)r * VCO);

    float bv = -__builtin_inff();
    int   bi = 0x7FFFFFFF;
    for (int c = lane; c < VCO / 4; c += 32) {    // 250 float4 chunks
        float4 q = src[c];
        int base = c * 4;
        combine_max(bv, bi, q.x, base + 0);
        combine_max(bv, bi, q.y, base + 1);
        combine_max(bv, bi, q.z, base + 2);
        combine_max(bv, bi, q.w, base + 3);
    }
#pragma unroll
    for (int m = 16; m; m >>= 1) {
        float ov = __shfl_xor(bv, m, 32);
        int   oi = __shfl_xor(bi, m, 32);
        combine_max(bv, bi, ov, oi);
    }
    // every lane now holds the row argmax; lanes 0..9 write the 10 beam copies
    int* dst = half ? out_ul : out_lr;
    if (lane < K_) dst[b * (K_ * S_) + lane * S_ + s] = bi;
}

// ---------------------------------------------------------------------------
// Kernel 3: pred_results select.
// ---------------------------------------------------------------------------
__global__ __launch_bounds__(256) void assemble_kernel(const int* __restrict__ struc,
                                                       const int* __restrict__ sc,
                                                       const int* __restrict__ lr,
                                                       const int* __restrict__ ul,
                                                       int* __restrict__ out) {
    const int idx = blockIdx.x * 256 + threadIdx.x;
    if (idx >= B_ * K_ * S_) return;
    const int b   = idx / (K_ * S_);
    const int rem = idx - b * (K_ * S_);
    const int j   = rem / S_;
    const int s   = rem - j * S_;
    const int st  = struc[b];
    int v = 0;
    if (st == 0)      v = (s == 0) ? sc[b * K_ + j] : 0;
    else if (st == 1) v = lr[idx];
    else if (st == 2) v = ul[idx];
    out[idx] = v;
}

extern "C" void kernel_launch(void* const* d_in, const int* in_sizes, int n_in,
                              void* d_out, int out_size, void* d_ws, size_t ws_size,
                              hipStream_t stream) {
    const int*   struc = (const int*)d_in[0];
    const float* sc    = (const float*)d_in[1];
    const float* lr    = (const float*)d_in[2];
    const float* ul    = (const float*)d_in[3];

    int* out   = (int*)d_out;
    int* o_sc  = out;                       // 512*10
    int* o_lr  = o_sc + B_ * K_;            // 512*10*40
    int* o_ul  = o_lr + B_ * K_ * S_;       // 512*10*40
    int* o_res = o_ul + B_ * K_ * S_;       // 512*10*40

    topk_sc_kernel<<<B_, 256, 0, stream>>>(sc, o_sc);
    argmax_seq_kernel<<<(2 * B_ * S_) / 8, 256, 0, stream>>>(lr, ul, o_lr, o_ul);
    assemble_kernel<<<(B_ * K_ * S_ + 255) / 256, 256, 0, stream>>>(struc, o_sc, o_lr, o_ul, o_res);
}